// DSAFTRankLoss_7799660609714
// MI455X (gfx1250) — compile-verified
//
#include <hip/hip_runtime.h>
#include <hip/hip_bf16.h>
#include <math.h>

typedef float v2f __attribute__((ext_vector_type(2)));
typedef float v8f __attribute__((ext_vector_type(8)));

#define ALPHA 0.001f
#define BETA  0.001f
#define EPSV  1e-32f
#define CHUNK 2048      // j-values staged in LDS per block (8 KB)
#define ITILE 128       // i-values per block (8 waves x 16 rows)

// ---------------------------------------------------------------- zero output
__global__ void zero_kernel(float* out) { out[0] = 0.0f; }

// ------------------------------------------- e[] + regularization penalties
__global__ void __launch_bounds__(256)
prep_kernel(const float* __restrict__ log_h, const float* __restrict__ dur,
            float* __restrict__ e, float* __restrict__ out, int n) {
    const int i = blockIdx.x * 256 + threadIdx.x;
    const float lh = log_h[i];
    const float ev = logf(dur[i] + EPSV) - lh;   // e_i = log(Y+eps) - theta
    e[i] = ev;
    const float inv_n = 1.0f / (float)n;
    float p = (ALPHA * ev * ev + BETA * lh * lh) * inv_n;

    __shared__ float red[256];
    red[threadIdx.x] = p;
    __syncthreads();
    #pragma unroll
    for (int s = 128; s > 0; s >>= 1) {
        if (threadIdx.x < s) red[threadIdx.x] += red[threadIdx.x + s];
        __syncthreads();
    }
    if (threadIdx.x == 0) atomicAdd(out, red[0]);
}

// --------------------------------------------------- pairwise rank loss core
// Each wave owns 16 i-rows. Per inner step a lane loads 4 contiguous e_j from
// LDS (b128), forms relu(e_j - e_i) into the 16x4 f32 A layout, and two
// v_wmma_f32_16x16x4_f32 (B = ones) accumulate row-sums. Two independent C
// accumulators break the serial D->C dependency chain so consecutive WMMAs
// can pipeline on the XDL unit. events[i] is applied after the j-sweep since
// it factors out of the inner sum.
__global__ void __launch_bounds__(256)
rank_loss_kernel(const float* __restrict__ e, const float* __restrict__ events,
                 float* __restrict__ out, int n, float inv_n2) {
    __shared__ __align__(16) float sj[CHUNK];

    const int tid  = threadIdx.x;
    const int wave = tid >> 5;
    const int lane = tid & 31;
    const int m    = lane & 15;    // row within 16-row tile
    const int h    = lane >> 4;    // lane half -> selects K slots {2h, 2h+1}

    // stage this block's j-chunk into LDS (coalesced, all 256 threads)
    const int j0 = blockIdx.y * CHUNK;
    #pragma unroll
    for (int t = tid; t < CHUNK; t += 256) sj[t] = e[j0 + t];
    __syncthreads();

    const int   i0 = blockIdx.x * ITILE + wave * 16;
    const float ei = e[i0 + m];

    v2f b; b.x = 1.0f; b.y = 1.0f;   // all-ones B (4x16)
    v8f c0 = {};                      // independent 16x16 f32 accumulators
    v8f c1 = {};

    #pragma unroll 4
    for (int jj = 0; jj < CHUNK; jj += 8) {
        const float4 v = *(const float4*)(&sj[jj + 4 * h]);
        v2f a0, a1;
        a0.x = fmaxf(v.x - ei, 0.0f);
        a0.y = fmaxf(v.y - ei, 0.0f);
        a1.x = fmaxf(v.z - ei, 0.0f);
        a1.y = fmaxf(v.w - ei, 0.0f);
        // D = A(16x4) * ones(4x16) + C   -> every column of D = row sums
        c0 = __builtin_amdgcn_wmma_f32_16x16x4_f32(false, a0, false, b,
                                                   (short)0, c0, false, false);
        c1 = __builtin_amdgcn_wmma_f32_16x16x4_f32(false, a1, false, b,
                                                   (short)0, c1, false, false);
    }

    // C layout: VGPR r holds row M=r (lanes 0-15) / M=8+r (lanes 16-31).
    float part = 0.0f;
    #pragma unroll
    for (int r = 0; r < 8; ++r)
        part += (c0[r] + c1[r]) * events[i0 + 8 * h + r];

    // all 16 lanes of a half hold identical values; one lane per half commits
    if ((lane & 15) == 0) atomicAdd(out, part * inv_n2);
}

// ----------------------------------------------------------------- launcher
extern "C" void kernel_launch(void* const* d_in, const int* in_sizes, int n_in,
                              void* d_out, int out_size, void* d_ws, size_t ws_size,
                              hipStream_t stream) {
    const float* log_h  = (const float*)d_in[0];
    const float* dur    = (const float*)d_in[1];
    const float* events = (const float*)d_in[2];
    float* out = (float*)d_out;
    float* e   = (float*)d_ws;           // n floats of scratch

    const int n = in_sizes[1];           // 16384
    const float inv_n2 = (float)(1.0 / ((double)n * (double)n));

    zero_kernel<<<1, 1, 0, stream>>>(out);
    prep_kernel<<<n / 256, 256, 0, stream>>>(log_h, dur, e, out, n);
    dim3 grid(n / ITILE, n / CHUNK);     // 128 x 8
    rank_loss_kernel<<<grid, 256, 0, stream>>>(e, events, out, n, inv_n2);
}